// EdgeEmbedding_49452253446963
// MI455X (gfx1250) — compile-verified
//
#include <hip/hip_runtime.h>

typedef __attribute__((ext_vector_type(2))) float v2f;
typedef __attribute__((ext_vector_type(8))) float v8f;

#define TED        64
#define NBASIS     8
#define R_MAX_F    6.0f
#define MLP_NORM_F 0.3535533905932738f   /* 1/sqrt(8) */

// One wave handles 16 edges per group iteration, TRANSPOSED GEMM:
//   D_t(16cols x 16edges) = W_t^T(16x8) @ basis^T(8x16)
// The f32 16x4 A-layout and 4x16 B-layout stripe identically, so the
// weight registers feed the A operand and the per-edge basis feeds B.
// Result: each lane owns ONE edge and 8 consecutive output columns per
// tile -> no cross-lane shuffles, b128 LDS gathers, b128 global stores.
__global__ void __launch_bounds__(256)
EdgeEmbedding_kernel(const int*   __restrict__ atom_types,   // [nNodes]
                     const int*   __restrict__ edge_index,   // [2, nEdges]
                     const float* __restrict__ edge_length,  // [nEdges]
                     const float* __restrict__ type_emb,     // [2,4,32] = 256 f32
                     const float* __restrict__ bessel_w,     // [8]
                     const float* __restrict__ mlp_w,        // [8,64]
                     float*       __restrict__ out,          // [nEdges, 64]
                     int nEdges)
{
    __shared__ __align__(16) float embLds[256];   // [table(2)][type(4)][col(32)]
    const int tid = threadIdx.x;
    if (tid < 256) embLds[tid] = type_emb[tid];
    __syncthreads();

    const int lane = tid & 31;
    const int n    = lane & 15;   // edge slot within group (N of D-tile)
    const int kh   = lane >> 4;   // K-half selector

    // ---- Loop-invariant weight tiles (mlp_w), now the A operand.
    // 16x4 f32 A layout: lanes 0-15 hold K={0,1} in VGPR{0,1}; lanes 16-31 K={2,3}.
    v2f Wlo[4], Whi[4];
#pragma unroll
    for (int t = 0; t < 4; ++t) {
        const int col = t * 16 + n;               // M slot = column of mlp_w
        Wlo[t].x = mlp_w[(2 * kh + 0) * TED + col];
        Wlo[t].y = mlp_w[(2 * kh + 1) * TED + col];
        Whi[t].x = mlp_w[(4 + 2 * kh + 0) * TED + col];
        Whi[t].y = mlp_w[(4 + 2 * kh + 1) * TED + col];
    }
    const float w0 = bessel_w[2 * kh + 0];
    const float w1 = bessel_w[2 * kh + 1];
    const float w2 = bessel_w[4 + 2 * kh + 0];
    const float w3 = bessel_w[4 + 2 * kh + 1];

    const int numFull = nEdges >> 4;             // full 16-edge groups
    const int waveId  = (int)((blockIdx.x * blockDim.x + tid) >> 5);
    const int nWaves  = (int)((gridDim.x * blockDim.x) >> 5);

    for (int g = waveId; g < numFull; g += nWaves) {
        const int base = g << 4;
        const int e    = base + n;               // this lane's edge
        const float r  = edge_length[e];
        const int  i0  = edge_index[e];
        const int  i1  = edge_index[nEdges + e];
        const int tyc  = atom_types[i0] & 3;
        const int tyn  = atom_types[i1] & 3;

        // Polynomial cutoff, p = 6:  1 - 28 x^6 + 48 x^7 - 21 x^8
        const float x  = r * (1.0f / R_MAX_F);
        const float x2 = x * x;
        const float x6 = x2 * x2 * x2;
        float cut = 1.0f - 28.0f * x6 + 48.0f * x6 * x - 21.0f * x6 * x2;
        cut = (x < 1.0f) ? cut : 0.0f;

        // 1/r via v_rcp_f32 + one Newton step (~0.5 ulp), avoids the
        // 12-instruction IEEE divide sequence.
        float inv = __builtin_amdgcn_rcpf(r);
        inv = inv + inv * __builtin_fmaf(-r, inv, 1.0f);
        const float s = cut * (MLP_NORM_F * 2.0f / R_MAX_F) * inv;

        // basis^T as the B operand (4x16 f32 B layout stripes the same way):
        // lanes 0-15 -> K{0,1}; lanes 16-31 -> K{2,3}; N = lane%16 = edge slot.
        v2f Alo, Ahi;
        Alo.x = s * __sinf(w0 * x);
        Alo.y = s * __sinf(w1 * x);
        Ahi.x = s * __sinf(w2 * x);
        Ahi.y = s * __sinf(w3 * x);

        // 4 independent WMMAs (K=0..3), then 4 dependent accumulates (K=4..7).
        v8f acc[4];
#pragma unroll
        for (int t = 0; t < 4; ++t) {
            v8f z = {};
            acc[t] = __builtin_amdgcn_wmma_f32_16x16x4_f32(
                false, Wlo[t], false, Alo, (short)0, z, false, false);
        }
#pragma unroll
        for (int t = 0; t < 4; ++t) {
            acc[t] = __builtin_amdgcn_wmma_f32_16x16x4_f32(
                false, Whi[t], false, Ahi, (short)0, acc[t], false, false);
        }

        // acc[t][v] = out column (16t + 8kh + v) of edge (base+n).
        // Lane-contiguous columns -> float4 LDS gathers + float4 stores.
        float* pe = out + (size_t)e * TED + (kh << 3);
        const float* tabc = embLds +       tyc * 32 + (kh << 3);
        const float* tabn = embLds + 128 + tyn * 32 + (kh << 3);
#pragma unroll
        for (int t = 0; t < 4; ++t) {
            const float* tb = (t < 2) ? tabc : tabn;
            const int   cb  = 16 * (t & 1);
            const float4 T0 = *(const float4*)(tb + cb);
            const float4 T1 = *(const float4*)(tb + cb + 4);
            float4 o0, o1;
            o0.x = acc[t][0] * T0.x;  o0.y = acc[t][1] * T0.y;
            o0.z = acc[t][2] * T0.z;  o0.w = acc[t][3] * T0.w;
            o1.x = acc[t][4] * T1.x;  o1.y = acc[t][5] * T1.y;
            o1.z = acc[t][6] * T1.z;  o1.w = acc[t][7] * T1.w;
            *(float4*)(pe + 16 * t)     = o0;
            *(float4*)(pe + 16 * t + 4) = o1;
        }
    }

    // -------- Tail (nEdges % 16 != 0): rare, scalar path, one wave. --------
    const int rem = nEdges & 15;
    if (rem && waveId == 0) {
        const int start = nEdges & ~15;
        for (int idx = lane; idx < rem * TED; idx += 32) {
            const int e   = start + (idx >> 6);
            const int col = idx & 63;
            const float r = edge_length[e];
            const float x = r * (1.0f / R_MAX_F);
            const float x2 = x * x;
            const float x6 = x2 * x2 * x2;
            float cut = 1.0f - 28.0f * x6 + 48.0f * x6 * x - 21.0f * x6 * x2;
            cut = (x < 1.0f) ? cut : 0.0f;
            const float s = cut * (MLP_NORM_F * 2.0f / R_MAX_F) / r;
            float d = 0.0f;
            for (int k = 0; k < NBASIS; ++k)
                d += s * __sinf(bessel_w[k] * x) * mlp_w[k * TED + col];
            const int tab = col >> 5;
            const int ty  = atom_types[edge_index[tab * nEdges + e]] & 3;
            out[(size_t)e * TED + col] = d * embLds[tab * 128 + ty * 32 + (col & 31)];
        }
    }
}

extern "C" void kernel_launch(void* const* d_in, const int* in_sizes, int n_in,
                              void* d_out, int out_size, void* d_ws, size_t ws_size,
                              hipStream_t stream) {
    (void)n_in; (void)d_ws; (void)ws_size; (void)out_size;
    const int*   atom_types  = (const int*)  d_in[0];
    const int*   edge_index  = (const int*)  d_in[1];
    const float* edge_length = (const float*)d_in[2];
    const float* type_emb    = (const float*)d_in[3];
    const float* bessel_w    = (const float*)d_in[4];
    const float* mlp_w       = (const float*)d_in[5];
    float*       out         = (float*)d_out;

    const int nEdges = in_sizes[2];
    const int numGroups = (nEdges + 15) / 16;

    // 8 waves/block; grid-stride over groups so loop-invariant weights amortize.
    int blocks = (numGroups + 7) / 8;
    if (blocks > 4096) blocks = 4096;
    if (blocks < 1)    blocks = 1;

    EdgeEmbedding_kernel<<<blocks, 256, 0, stream>>>(
        atom_types, edge_index, edge_length, type_emb, bessel_w, mlp_w,
        out, nEdges);
}